// HESLoss_86225763435426
// MI455X (gfx1250) — compile-verified
//
#include <hip/hip_runtime.h>
#include <math.h>

typedef __attribute__((ext_vector_type(2))) float v2f;
typedef __attribute__((ext_vector_type(8))) float v8f;
typedef __attribute__((ext_vector_type(2))) int   v2i_t;

#define B_MOLE   4096
#define NROW     8192      // 2B rows in combined embedding
#define DIM      256
#define N_ATOMS  131072
#define N_MOTIFS 32768
#define BROW     64        // output rows per workgroup (4 waves x 16)
#define BCOL     64        // columns per iteration
#define NSPLIT   8         // column splits (flash-style two-stage softmax)
#define SPLIT_W  (NROW / NSPLIT)   // 1024 columns per split
#define LDSS     260       // padded row stride (floats): bank-conflict-free, 8B aligned
#define TAU_INV  10.0f
// (EPS - 1e-8)^4 : mask test "sqrt(sqrt(d2))+1e-8 < EPS" <=> "d2 < this"
#define D2_THR   6.249995e-06f

#if defined(__has_builtin)
#if __has_builtin(__builtin_amdgcn_global_load_async_to_lds_b64)
#define HAVE_ASYNC_LDS 1
#endif
#endif

typedef __attribute__((address_space(1))) v2i_t* g_b64_ptr;  // global v2i*
typedef __attribute__((address_space(3))) v2i_t* l_b64_ptr;  // LDS v2i*

__device__ __forceinline__ void wait_asynccnt0() {
#if defined(__has_builtin)
#if __has_builtin(__builtin_amdgcn_s_wait_asynccnt)
  __builtin_amdgcn_s_wait_asynccnt(0);
  return;
#else
  asm volatile("s_wait_asynccnt 0" ::: "memory");
  return;
#endif
#else
  asm volatile("s_wait_asynccnt 0" ::: "memory");
#endif
}

// ---------- reduction helpers (wave32) ----------
__device__ __forceinline__ float red_sum32(float v) {
#pragma unroll
  for (int s = 1; s <= 16; s <<= 1) v += __shfl_xor(v, s, 32);
  return v;
}

// cooperative copy: rows x 256 floats -> LDS with padded stride (128 threads)
__device__ __forceinline__ void copy_panel(const float* __restrict__ src,
                                           float* lds, int rows, int tid) {
#if HAVE_ASYNC_LDS
  for (int idx = tid; idx < rows * (DIM / 2); idx += 128) {
    int r = idx >> 7, c = idx & 127;
    float* gp = const_cast<float*>(src) + (size_t)r * DIM + 2 * c;
    float* lp = lds + r * LDSS + 2 * c;
    __builtin_amdgcn_global_load_async_to_lds_b64(
        (g_b64_ptr)(v2i_t*)gp, (l_b64_ptr)(v2i_t*)lp, 0, 0);
  }
  wait_asynccnt0();
#else
  for (int idx = tid; idx < rows * (DIM / 2); idx += 128) {
    int r = idx >> 7, c = idx & 127;
    float2 v = ((const float2*)(src + (size_t)r * DIM))[c];
    lds[r * LDSS + 2 * c]     = v.x;
    lds[r * LDSS + 2 * c + 1] = v.y;
  }
#endif
}

// ---------- 1) row-normalize combined embedding: e = x / max(||x||, 1e-12) ----------
__global__ __launch_bounds__(256) void hes_normalize(const float* __restrict__ g,
                                                     const float* __restrict__ sc,
                                                     float* __restrict__ e) {
  int row  = blockIdx.x * 8 + (threadIdx.x >> 5);
  int lane = threadIdx.x & 31;
  const float* src = (row < B_MOLE) ? (g + (size_t)row * DIM)
                                    : (sc + (size_t)(row - B_MOLE) * DIM);
  float v[8], ss = 0.f;
#pragma unroll
  for (int q = 0; q < 8; ++q) { v[q] = src[lane + 32 * q]; ss += v[q] * v[q]; }
  ss = red_sum32(ss);
  float inv = 1.0f / fmaxf(sqrtf(ss), 1e-12f);
  float* dst = e + (size_t)row * DIM;
#pragma unroll
  for (int q = 0; q < 8; ++q) dst[lane + 32 * q] = v[q] * inv;
}

// ---------- 2) fused GEMM (v_wmma_f32_16x16x4_f32) + online log-softmax + sparse mask ----------
// grid = (NROW/BROW, NSPLIT); each workgroup covers 64 rows x 1024 columns and
// writes per-row partials {max, sumexp, masked_sum, masked_cnt} to a unique slot.
__global__ __launch_bounds__(128) void hes_contrastive(const float* __restrict__ e,
                                                       const float* __restrict__ props, // prop_pred [4096][4]
                                                       float4* __restrict__ part) {
  __shared__ __align__(16) float ldsA[BROW * LDSS];
  __shared__ __align__(16) float ldsB[BCOL * LDSS];
  __shared__ __align__(16) float ldsPj[BCOL * 4];

  const int tid  = threadIdx.x;
  const int lane = tid & 31;
  const int wave = tid >> 5;     // 0..3
  const int hh   = lane >> 4;    // half-wave (0/1): K pair select + row-half of C
  const int l16  = lane & 15;
  const int rowBase  = blockIdx.x * BROW;
  const int split    = blockIdx.y;
  const int colStart = split * SPLIT_W;

  // stage A panel (64 x 256) once
  copy_panel(e + (size_t)rowBase * DIM, ldsA, BROW, tid);

  // per-lane row properties: 8 rows (WMMA C layout: lanes<16 -> M 0..7, lanes>=16 -> M 8..15)
  float pi[8][4];
#pragma unroll
  for (int r = 0; r < 8; ++r) {
    int gi = rowBase + wave * 16 + r + 8 * hh;
    float4 p = *(const float4*)(props + (size_t)(gi & (B_MOLE - 1)) * 4);
    pi[r][0] = p.x; pi[r][1] = p.y; pi[r][2] = p.z; pi[r][3] = p.w;
  }

  // lane-local online-softmax state (each lane owns 4 columns per block)
  float run_max[8], run_sum[8], msum[8], mcnt[8];
#pragma unroll
  for (int r = 0; r < 8; ++r) { run_max[r] = -3.0e38f; run_sum[r] = 0.f; msum[r] = 0.f; mcnt[r] = 0.f; }

  for (int colBase = colStart; colBase < colStart + SPLIT_W; colBase += BCOL) {
    __syncthreads();   // protect previous-iteration LDS reads
    copy_panel(e + (size_t)colBase * DIM, ldsB, BCOL, tid);
    if (tid < BCOL) {
      float4 p = *(const float4*)(props + (size_t)((colBase + tid) & (B_MOLE - 1)) * 4);
      ((float4*)ldsPj)[tid] = p;
    }
    __syncthreads();

    v8f a0 = {}, a1 = {}, a2 = {}, a3 = {};
    const float* Ap = ldsA + (wave * 16 + l16) * LDSS + 2 * hh;
    const float* B0 = ldsB + ( 0 + l16) * LDSS + 2 * hh;
    const float* B1 = ldsB + (16 + l16) * LDSS + 2 * hh;
    const float* B2 = ldsB + (32 + l16) * LDSS + 2 * hh;
    const float* B3 = ldsB + (48 + l16) * LDSS + 2 * hh;
#pragma unroll 4
    for (int k0 = 0; k0 < DIM; k0 += 4) {
      v2f a  = *(const v2f*)(Ap + k0);
      v2f b0 = *(const v2f*)(B0 + k0);
      v2f b1 = *(const v2f*)(B1 + k0);
      v2f b2 = *(const v2f*)(B2 + k0);
      v2f b3 = *(const v2f*)(B3 + k0);
      a0 = __builtin_amdgcn_wmma_f32_16x16x4_f32(false, a, false, b0, (short)0, a0, false, false);
      a1 = __builtin_amdgcn_wmma_f32_16x16x4_f32(false, a, false, b1, (short)0, a1, false, false);
      a2 = __builtin_amdgcn_wmma_f32_16x16x4_f32(false, a, false, b2, (short)0, a2, false, false);
      a3 = __builtin_amdgcn_wmma_f32_16x16x4_f32(false, a, false, b3, (short)0, a3, false, false);
    }

    // hoist column properties for this lane's 4 columns (invariant over r)
    float pj[4][4];
#pragma unroll
    for (int t = 0; t < 4; ++t) {
      const float* p = &ldsPj[(t * 16 + l16) * 4];
      pj[t][0] = p[0]; pj[t][1] = p[1]; pj[t][2] = p[2]; pj[t][3] = p[3];
    }

    // lane-local online softmax update + masked accumulation (no cross-lane work here)
#pragma unroll
    for (int r = 0; r < 8; ++r) {
      float v0 = a0[r] * TAU_INV, v1 = a1[r] * TAU_INV;
      float v2_ = a2[r] * TAU_INV, v3 = a3[r] * TAU_INV;
      float cm = fmaxf(fmaxf(v0, v1), fmaxf(v2_, v3));
      float nm = fmaxf(run_max[r], cm);
      run_sum[r] = run_sum[r] * __expf(run_max[r] - nm)
                 + __expf(v0 - nm) + __expf(v1 - nm)
                 + __expf(v2_ - nm) + __expf(v3 - nm);
      run_max[r] = nm;

      int gi = rowBase + wave * 16 + r + 8 * hh;
      float vr[4] = { v0, v1, v2_, v3 };
#pragma unroll
      for (int t = 0; t < 4; ++t) {
        int gj = colBase + t * 16 + l16;
        float d0 = pi[r][0] - pj[t][0], d1 = pi[r][1] - pj[t][1];
        float d2 = pi[r][2] - pj[t][2], d3 = pi[r][3] - pj[t][3];
        float q  = d0 * d0 + d1 * d1 + d2 * d2 + d3 * d3;
        bool m = (q < D2_THR) && (gi != gj);
        msum[r] += m ? vr[t] : 0.0f;
        mcnt[r] += m ? 1.0f  : 0.0f;
      }
    }
  }

  // one-time cross-lane combine over the 16 lanes of each half-wave
#pragma unroll
  for (int r = 0; r < 8; ++r) {
#pragma unroll
    for (int s = 1; s <= 8; s <<= 1) {
      float om = __shfl_xor(run_max[r], s, 32);
      float os = __shfl_xor(run_sum[r], s, 32);
      float nm = fmaxf(run_max[r], om);
      run_sum[r] = run_sum[r] * __expf(run_max[r] - nm) + os * __expf(om - nm);
      run_max[r] = nm;
      msum[r] += __shfl_xor(msum[r], s, 32);
      mcnt[r] += __shfl_xor(mcnt[r], s, 32);
    }
  }
  if (l16 == 0) {
#pragma unroll
    for (int r = 0; r < 8; ++r) {
      int gi = rowBase + wave * 16 + r + 8 * hh;
      part[(size_t)gi * NSPLIT + split] =
          make_float4(run_max[r], run_sum[r], msum[r], mcnt[r]);
    }
  }
}

// ---------- 2b) merge per-row split partials -> contrastive loss ----------
__global__ __launch_bounds__(256) void hes_cmerge(const float4* __restrict__ part,
                                                  float* __restrict__ acc) {
  int i = blockIdx.x * 256 + threadIdx.x;   // row 0..8191
  float m = -3.0e38f, s = 0.f, ms = 0.f, mc = 0.f;
#pragma unroll
  for (int p = 0; p < NSPLIT; ++p) {
    float4 q = part[(size_t)i * NSPLIT + p];
    float nm = fmaxf(m, q.x);
    s = s * __expf(m - nm) + q.y * __expf(q.x - nm);
    m = nm;
    ms += q.z; mc += q.w;
  }
  float lse = m + __logf(s);
  float rl  = -(ms - mc * lse) / fmaxf(mc, 1.0f);
  float lsum = (mc > 0.f) ? rl : 0.0f;
  float vcnt = (mc > 0.f) ? 1.0f : 0.0f;
  lsum = red_sum32(lsum);
  vcnt = red_sum32(vcnt);
  if ((threadIdx.x & 31) == 0) {
    atomicAdd(&acc[0], lsum);
    atomicAdd(&acc[1], vcnt);
  }
}

// ---------- 3) segment sums (wave per element row; coalesced per-lane atomics) ----------
__global__ __launch_bounds__(256) void hes_segsum(const float* __restrict__ emb,
                                                  const int* __restrict__ batch, int n,
                                                  float* __restrict__ sums,
                                                  float* __restrict__ cnts,
                                                  int* __restrict__ bmax, int trackMax) {
  int a = blockIdx.x * 8 + (threadIdx.x >> 5);
  if (a >= n) return;
  int lane = threadIdx.x & 31;
  int b = batch[a];
  const float* src = emb + (size_t)a * DIM;
  float* dst = sums + (size_t)b * DIM;
#pragma unroll
  for (int q = 0; q < 8; ++q) atomicAdd(&dst[lane + 32 * q], src[lane + 32 * q]);
  if (lane == 0) {
    atomicAdd(&cnts[b], 1.0f);
    if (trackMax) atomicMax(bmax, b);
  }
}

// ---------- 4) fragment alignment: ||mean_g - mean_s|| over valid segments ----------
__global__ __launch_bounds__(256) void hes_fragalign(const float* __restrict__ sumg,
                                                     const float* __restrict__ cntg,
                                                     const float* __restrict__ sums,
                                                     const float* __restrict__ cnts,
                                                     float* __restrict__ acc) {
  int b = blockIdx.x * 8 + (threadIdx.x >> 5);
  int lane = threadIdx.x & 31;
  float cg = cntg[b], cs = cnts[b];
  float ig = 1.0f / fmaxf(cg, 1.0f), is = 1.0f / fmaxf(cs, 1.0f);
  float ss = 0.f;
#pragma unroll
  for (int q = 0; q < 8; ++q) {
    size_t o = (size_t)b * DIM + lane + 32 * q;
    float d = sumg[o] * ig - sums[o] * is;
    ss += d * d;
  }
  ss = red_sum32(ss);
  if (lane == 0 && cg > 0.f && cs > 0.f) atomicAdd(&acc[2], sqrtf(ss));
}

// ---------- 5) shape alignment: mean row-norm of (emb_sc_shape - emb_shape) ----------
__global__ __launch_bounds__(256) void hes_scalign(const float* __restrict__ x,
                                                   const float* __restrict__ y,
                                                   float* __restrict__ acc) {
  int row = blockIdx.x * 8 + (threadIdx.x >> 5);
  int lane = threadIdx.x & 31;
  float ss = 0.f;
#pragma unroll
  for (int q = 0; q < 8; ++q) {
    size_t o = (size_t)row * DIM + lane + 32 * q;
    float d = x[o] - y[o];
    ss += d * d;
  }
  ss = red_sum32(ss);
  if (lane == 0) atomicAdd(&acc[3], sqrtf(ss));
}

// ---------- 6) property MSE ----------
__global__ __launch_bounds__(256) void hes_propmse(const float* __restrict__ p,
                                                   const float* __restrict__ t,
                                                   float* __restrict__ acc) {
  int i = blockIdx.x * 256 + threadIdx.x;
  float d = p[i] - t[i];
  float v = red_sum32(d * d);
  if ((threadIdx.x & 31) == 0) atomicAdd(&acc[4], v);
}

// ---------- 7) finalize the six outputs ----------
__global__ void hes_finalize(const float* __restrict__ acc, const int* __restrict__ bmax,
                             float* __restrict__ out) {
  float l_contr   = (acc[1] > 0.f) ? acc[0] / fmaxf(acc[1], 1.0f) : 0.0f;
  float l_local_g = acc[2] / ((float)(*bmax) + 1.0f);
  float l_sc_g    = acc[3] / (float)N_MOTIFS;
  float l_prop    = acc[4] / (float)(B_MOLE * 4);
  out[0] = 1.0f * l_local_g + 0.8f * l_contr + 1.0f * l_sc_g + 0.8f * l_contr + 0.5f * l_prop;
  out[1] = l_local_g;
  out[2] = l_contr;
  out[3] = l_sc_g;
  out[4] = l_contr;
  out[5] = l_prop;
}

extern "C" void kernel_launch(void* const* d_in, const int* in_sizes, int n_in,
                              void* d_out, int out_size, void* d_ws, size_t ws_size,
                              hipStream_t stream) {
  const float* emb_g_mole  = (const float*)d_in[0];
  const float* emb_g_frag  = (const float*)d_in[1];
  const float* emb_sc_mole = (const float*)d_in[2];
  const float* emb_sc_shape= (const float*)d_in[3];
  const float* emb_motif   = (const float*)d_in[4];
  const float* emb_shape   = (const float*)d_in[5];
  const float* prop_pred   = (const float*)d_in[6];
  const float* prop_target = (const float*)d_in[7];
  const int*   batch_g     = (const int*)d_in[8];
  const int*   batch_sc    = (const int*)d_in[9];
  float* out = (float*)d_out;

  // workspace layout
  float*  ws_acc  = (float*)d_ws;                                  // acc[0..4]
  int*    ws_bmax = (int*)((char*)d_ws + 32);
  float*  ws_sumg = (float*)((char*)d_ws + 64);                    // [4096][256]
  float*  ws_sums = ws_sumg + (size_t)B_MOLE * DIM;                // [4096][256]
  float*  ws_cntg = ws_sums + (size_t)B_MOLE * DIM;                // [4096]
  float*  ws_cnts = ws_cntg + B_MOLE;                              // [4096]
  float*  ws_e    = ws_cnts + B_MOLE;                              // [8192][256]
  float4* ws_part = (float4*)(ws_e + (size_t)NROW * DIM);          // [8192][NSPLIT]
  size_t zeroBytes = 64 + (size_t)2 * B_MOLE * DIM * 4 + (size_t)2 * B_MOLE * 4;

  (void)hipMemsetAsync(d_ws, 0, zeroBytes, stream);

  hes_normalize<<<NROW / 8, 256, 0, stream>>>(emb_g_mole, emb_sc_mole, ws_e);
  hes_contrastive<<<dim3(NROW / BROW, NSPLIT), 128, 0, stream>>>(ws_e, prop_pred, ws_part);
  hes_cmerge<<<NROW / 256, 256, 0, stream>>>(ws_part, ws_acc);

  hes_segsum<<<N_ATOMS / 8, 256, 0, stream>>>(emb_g_frag, batch_g, N_ATOMS,
                                              ws_sumg, ws_cntg, ws_bmax, 1);
  hes_segsum<<<N_MOTIFS / 8, 256, 0, stream>>>(emb_motif, batch_sc, N_MOTIFS,
                                               ws_sums, ws_cnts, ws_bmax, 0);
  hes_fragalign<<<B_MOLE / 8, 256, 0, stream>>>(ws_sumg, ws_cntg, ws_sums, ws_cnts, ws_acc);
  hes_scalign<<<N_MOTIFS / 8, 256, 0, stream>>>(emb_sc_shape, emb_shape, ws_acc);
  hes_propmse<<<(B_MOLE * 4) / 256, 256, 0, stream>>>(prop_pred, prop_target, ws_acc);
  hes_finalize<<<1, 1, 0, stream>>>(ws_acc, ws_bmax, out);
}